// SpectralFeatureExtractor_91336774516757
// MI455X (gfx1250) — compile-verified
//
#include <hip/hip_runtime.h>

#define N_POINTS 1000
#define FREQ_STEP (500.0f / 999.0f)   // linspace(0, 500, 1000) step

typedef __attribute__((ext_vector_type(2))) float f32x2;
typedef __attribute__((ext_vector_type(8))) float f32x8;

// One wave = 16 rows. WMMA f32 16x16x4 accumulates S0=sum(s), S1=sum(f*s),
// S2=sum(f^2*s) into D columns 0..2. Pass 1 (HBM-bound) does only WMMA +
// v_max_f32. Pass 2 (WGP$/L2-hot, non-temporal) does the threshold count and
// first-occurrence argmax against the known max.
__global__ __launch_bounds__(256) void spectral_features_kernel(
    const float* __restrict__ spectra, float* __restrict__ out, int n_rows)
{
    const int lane  = threadIdx.x & 31;
    const int wave  = threadIdx.x >> 5;
    const int gwave = blockIdx.x * 8 + wave;
    const int row_base = gwave * 16;

    const int r   = lane & 15;               // row index (0..15) this lane serves
    const int row = row_base + r;
    const bool row_valid = row < n_rows;
    const int crow = row_valid ? row : (n_rows - 1);   // clamp: keep EXEC all-1s for WMMA

    const int delta = (lane < 16) ? 0 : 2;   // K offset per ISA A-matrix 16x4 layout
    const float* base = spectra + (size_t)crow * N_POINTS + delta;

    // B-matrix lane constants: column n of B holds [1, f, f^2, 0...].
    const float Ac = (r == 0) ? 1.0f : 0.0f;
    const float Bc = (r == 1) ? 1.0f : 0.0f;
    const float Cc = (r == 2) ? 1.0f : 0.0f;

    // Second-difference recurrence: b(f) = Ac + Bc*f + Cc*f^2, f advances by D=4*step.
    const float D  = 4.0f * FREQ_STEP;
    const float f0 = (float)(delta)     * FREQ_STEP;
    const float f1 = (float)(delta + 1) * FREQ_STEP;
    f32x2 b, db;
    b.x  = Ac + f0 * (Bc + Cc * f0);
    b.y  = Ac + f1 * (Bc + Cc * f1);
    db.x = Bc * D + Cc * D * (2.0f * f0 + D);
    db.y = Bc * D + Cc * D * (2.0f * f1 + D);
    const float c2 = 2.0f * Cc * D * D;

    f32x8 acc = {};                  // D/C accumulator: 16x16 f32, cols 0..2 used
    float m = -__builtin_inff();     // per-lane running max

    // ---- Pass 1: HBM stream. Load b64 -> WMMA + max only. ----
    for (int k = 0; k < N_POINTS; k += 4) {
        f32x2 a = *(const f32x2*)(base + k);        // A: row r, K = delta, delta+1

        acc = __builtin_amdgcn_wmma_f32_16x16x4_f32(
                  false, a, false, b, (short)0, acc, false, false);

        m = fmaxf(m, a.x);
        m = fmaxf(m, a.y);

        b.x += db.x;  db.x += c2;                   // advance B to next K-chunk
        b.y += db.y;  db.y += c2;
    }

    // Combine max across the lane pair (l, l+16) that shares a row.
    m = fmaxf(m, __shfl_xor(m, 16, 32));

    // ---- Pass 2: cache-hot, non-temporal. Count above threshold + argmax. ----
    const float thr = 0.1f * m;
    int cnt = 0;
    int mi  = 0x7fffffff;
    int i0  = delta;                                // index of a.x this iteration
    for (int k = 0; k < N_POINTS; k += 4) {
        f32x2 a = __builtin_nontemporal_load((const f32x2*)(base + k));
        cnt += (a.x > thr) ? 1 : 0;
        cnt += (a.y > thr) ? 1 : 0;
        if (a.x == m) mi = min(mi, i0);             // exact re-read: bitwise equal hit
        if (a.y == m) mi = min(mi, i0 + 1);         // increasing order -> first occurrence
        i0 += 4;
    }
    cnt += __shfl_xor(cnt, 16, 32);
    mi = min(mi, __shfl_xor(mi, 16, 32));

    // ---- Unpack S0/S1/S2 for row r from the accumulator. ----
    // D layout: row m<8 -> VGPR m, lane = col; row m>=8 -> VGPR m-8, lane = 16+col.
    float S0 = 0.0f, S1 = 0.0f, S2 = 0.0f;
    const int srcbase = (r < 8) ? 0 : 16;
    const int jj = r & 7;
#pragma unroll
    for (int j = 0; j < 8; ++j) {
        float a0 = __shfl(acc[j], srcbase + 0, 32);
        float a1 = __shfl(acc[j], srcbase + 1, 32);
        float a2 = __shfl(acc[j], srcbase + 2, 32);
        if (jj == j) { S0 = a0; S1 = a1; S2 = a2; }
    }

    if (lane < 16 && row_valid) {
        const float centroid = S1 / S0;
        float var = S2 / S0 - centroid * centroid;  // = sum((f-c)^2 s)/I
        var = var > 0.0f ? var : 0.0f;
        float* o = out + (size_t)row * 6;
        o[0] = S0;                                  // integral
        o[1] = m;                                   // peak height
        o[2] = (float)mi * FREQ_STEP;               // peak freq
        o[3] = centroid;
        o[4] = __builtin_sqrtf(var);                // std dev
        o[5] = (float)cnt;                          // above-threshold count
    }
}

extern "C" void kernel_launch(void* const* d_in, const int* in_sizes, int n_in,
                              void* d_out, int out_size, void* d_ws, size_t ws_size,
                              hipStream_t stream) {
    (void)n_in; (void)out_size; (void)d_ws; (void)ws_size;
    const float* spectra = (const float*)d_in[0];
    float* out = (float*)d_out;
    const int n_rows = in_sizes[0] / N_POINTS;      // 131072
    const int rows_per_block = 128;                 // 8 waves x 16 rows
    const int blocks = (n_rows + rows_per_block - 1) / rows_per_block;
    spectral_features_kernel<<<blocks, 256, 0, stream>>>(spectra, out, n_rows);
}